// GNN_9689446219975
// MI455X (gfx1250) — compile-verified
//
#include <hip/hip_runtime.h>
#include <hip/hip_bf16.h>
#include <math.h>

typedef float v2f __attribute__((ext_vector_type(2)));
typedef float v8f __attribute__((ext_vector_type(8)));

#define IN_C 64
#define HID_C 64
#define OUT_C 32

// ---------- helpers: order-preserving float<->uint map for atomic max ----------
__device__ __forceinline__ unsigned fmap(float f) {
    unsigned u = __float_as_uint(f);
    return (u & 0x80000000u) ? ~u : (u | 0x80000000u);
}
__device__ __forceinline__ float funmap(unsigned u) {
    return (u & 0x80000000u) ? __uint_as_float(u & 0x7FFFFFFFu) : __uint_as_float(~u);
}
__device__ __forceinline__ float leaky(float v) { return v > 0.0f ? v : 0.2f * v; }

// ---------- K1: edge scatter for SAGE mean aggregation ----------
// thread t -> (edge, 4-channel chunk). 16 chunks per edge cover IN_C=64.
__global__ void k_agg(const float* __restrict__ x, const int* __restrict__ ei,
                      float* __restrict__ agg, float* __restrict__ deg, int nE) {
    int t = blockIdx.x * blockDim.x + threadIdx.x;
    int e = t >> 4;
    if (e >= nE) return;
    int ch = (t & 15) * 4;
    int s = ei[e];
    int d = ei[nE + e];
    const float4 v = *(const float4*)(x + (size_t)s * IN_C + ch);
    float* ap = agg + (size_t)d * IN_C + ch;
    atomicAdd(ap + 0, v.x);
    atomicAdd(ap + 1, v.y);
    atomicAdd(ap + 2, v.z);
    atomicAdd(ap + 3, v.w);
    if ((t & 15) == 0) atomicAdd(deg + d, 1.0f);
}

// ---------- K2: fused WMMA transforms ----------
// h = relu((agg/deg) @ W_l + b1 + x @ W_r); z = h @ W_g
// one wave per 16-row node tile; 8 waves (8 tiles = 128 rows) per block.
__global__ __launch_bounds__(256) void k_gemm(
    const float* __restrict__ x, const float* __restrict__ agg,
    const float* __restrict__ deg, const float* __restrict__ Wl,
    const float* __restrict__ Wr, const float* __restrict__ b1,
    const float* __restrict__ Wg, float* __restrict__ z, int n) {
    __shared__ float sWl[HID_C * IN_C];      // 16 KB (K-major: [k][col])
    __shared__ float sWr[HID_C * IN_C];      // 16 KB
    __shared__ float sWg[HID_C * OUT_C];     // 8 KB
    __shared__ float sB1[HID_C];
    __shared__ float sH[8][16 * HID_C];      // per-wave H tile, 32 KB

    for (int i = threadIdx.x; i < IN_C * HID_C; i += 256) { sWl[i] = Wl[i]; sWr[i] = Wr[i]; }
    for (int i = threadIdx.x; i < HID_C * OUT_C; i += 256) sWg[i] = Wg[i];
    if (threadIdx.x < HID_C) sB1[threadIdx.x] = b1[threadIdx.x];
    __syncthreads();

    const int wave = threadIdx.x >> 5;
    const int lane = threadIdx.x & 31;
    const int half = lane >> 4;      // selects K pair {2h, 2h+1} within K=4 step
    const int ln   = lane & 15;      // row (A) / col (B,C,D) index
    const int row0 = (blockIdx.x * 8 + wave) * 16;
    const int row  = row0 + ln;
    const bool rv  = row < n;

    float rinv = 1.0f;
    if (rv) rinv = 1.0f / fmaxf(deg[row], 1.0f);

    // Preload A fragments for all 16 K-steps of both input matrices.
    v2f aA[16], aX[16];
    const float* ar = agg + (size_t)row * IN_C;
    const float* xr = x + (size_t)row * IN_C;
#pragma unroll
    for (int kk = 0; kk < 16; ++kk) {
        int c = kk * 4 + half * 2;
        if (rv) {
            aA[kk].x = ar[c] * rinv;  aA[kk].y = ar[c + 1] * rinv;
            aX[kk].x = xr[c];         aX[kk].y = xr[c + 1];
        } else {
            aA[kk] = (v2f){0.0f, 0.0f};
            aX[kk] = (v2f){0.0f, 0.0f};
        }
    }

    // H = aggScaled @ W_l + x @ W_r + b1, 4 column tiles of 16.
    v8f hacc[4];
#pragma unroll
    for (int ct = 0; ct < 4; ++ct) {
        float bb = sB1[ct * 16 + ln];
        hacc[ct] = (v8f){bb, bb, bb, bb, bb, bb, bb, bb};
#pragma unroll
        for (int kk = 0; kk < 16; ++kk) {
            int krow = kk * 4 + half * 2;
            v2f b;
            b.x = sWl[krow * HID_C + ct * 16 + ln];
            b.y = sWl[(krow + 1) * HID_C + ct * 16 + ln];
            hacc[ct] = __builtin_amdgcn_wmma_f32_16x16x4_f32(
                false, aA[kk], false, b, (short)0, hacc[ct], false, false);
            b.x = sWr[krow * HID_C + ct * 16 + ln];
            b.y = sWr[(krow + 1) * HID_C + ct * 16 + ln];
            hacc[ct] = __builtin_amdgcn_wmma_f32_16x16x4_f32(
                false, aX[kk], false, b, (short)0, hacc[ct], false, false);
        }
    }

    // relu + park H in per-wave LDS so we can re-fragment it as an A operand.
#pragma unroll
    for (int ct = 0; ct < 4; ++ct)
#pragma unroll
        for (int r = 0; r < 8; ++r)
            sH[wave][(r + 8 * half) * HID_C + ct * 16 + ln] = fmaxf(hacc[ct][r], 0.0f);

    // A fragments of H (same wave for read/write, across lanes; DS waits inserted by compiler).
    v2f aH[16];
#pragma unroll
    for (int kk = 0; kk < 16; ++kk) {
        int c = kk * 4 + half * 2;
        aH[kk].x = sH[wave][ln * HID_C + c];
        aH[kk].y = sH[wave][ln * HID_C + c + 1];
    }

    // Z = H @ W_g, 2 column tiles of 16.
    v8f zacc[2];
#pragma unroll
    for (int zt = 0; zt < 2; ++zt) {
        zacc[zt] = (v8f){0.f, 0.f, 0.f, 0.f, 0.f, 0.f, 0.f, 0.f};
#pragma unroll
        for (int kk = 0; kk < 16; ++kk) {
            int krow = kk * 4 + half * 2;
            v2f b;
            b.x = sWg[krow * OUT_C + zt * 16 + ln];
            b.y = sWg[(krow + 1) * OUT_C + zt * 16 + ln];
            zacc[zt] = __builtin_amdgcn_wmma_f32_16x16x4_f32(
                false, aH[kk], false, b, (short)0, zacc[zt], false, false);
        }
    }

#pragma unroll
    for (int zt = 0; zt < 2; ++zt)
#pragma unroll
        for (int r = 0; r < 8; ++r) {
            int rr = row0 + r + 8 * half;
            if (rr < n) z[(size_t)rr * OUT_C + zt * 16 + ln] = zacc[zt][r];
        }
}

// ---------- K3: per-node attention scores + seed m with the self-loop ----------
__global__ void k_att(const float* __restrict__ z, const float* __restrict__ att_s,
                      const float* __restrict__ att_d, float* __restrict__ a_s,
                      float* __restrict__ a_d, unsigned* __restrict__ m_u, int n) {
    int i = blockIdx.x * blockDim.x + threadIdx.x;
    if (i >= n) return;
    const float* zr = z + (size_t)i * OUT_C;
    float ss = 0.0f, dd = 0.0f;
#pragma unroll
    for (int c = 0; c < OUT_C; ++c) { float v = zr[c]; ss += v * att_s[c]; dd += v * att_d[c]; }
    a_s[i] = ss;
    a_d[i] = dd;
    m_u[i] = fmap(leaky(ss + dd));  // self-loop edge always present
}

// ---------- K4: per-edge running max ----------
__global__ void k_emax(const int* __restrict__ ei, const float* __restrict__ a_s,
                       const float* __restrict__ a_d, unsigned* __restrict__ m_u, int nE) {
    int e = blockIdx.x * blockDim.x + threadIdx.x;
    if (e >= nE) return;
    int s = ei[e], d = ei[nE + e];
    atomicMax(m_u + d, fmap(leaky(a_s[s] + a_d[d])));
}

// ---------- K5: per-node prep: unmap m, seed denom/outacc with self-loop ----------
__global__ void k_prep(const unsigned* __restrict__ m_u, const float* __restrict__ a_s,
                       const float* __restrict__ a_d, const float* __restrict__ z,
                       float* __restrict__ mbuf, float* __restrict__ denom,
                       float* __restrict__ outacc, int n) {
    int i = blockIdx.x * blockDim.x + threadIdx.x;
    if (i >= n) return;
    float m = funmap(m_u[i]);
    mbuf[i] = m;
    float w = __expf(leaky(a_s[i] + a_d[i]) - m);
    denom[i] = w;
    const float* zr = z + (size_t)i * OUT_C;
    float* op = outacc + (size_t)i * OUT_C;
#pragma unroll
    for (int c = 0; c < OUT_C / 4; ++c) {
        float4 v = *(const float4*)(zr + c * 4);
        float4 o; o.x = w * v.x; o.y = w * v.y; o.z = w * v.z; o.w = w * v.w;
        *(float4*)(op + c * 4) = o;
    }
}

// ---------- K6: per-edge softmax-weighted scatter ----------
// thread t -> (edge, 4-channel chunk); 8 chunks cover OUT_C=32.
__global__ void k_eagg(const int* __restrict__ ei, const float* __restrict__ a_s,
                       const float* __restrict__ a_d, const float* __restrict__ mbuf,
                       const float* __restrict__ z, float* __restrict__ denom,
                       float* __restrict__ outacc, int nE) {
    int t = blockIdx.x * blockDim.x + threadIdx.x;
    int e = t >> 3;
    if (e >= nE) return;
    int ch = (t & 7) * 4;
    int s = ei[e], d = ei[nE + e];
    float ex = __expf(leaky(a_s[s] + a_d[d]) - mbuf[d]);
    if ((t & 7) == 0) atomicAdd(denom + d, ex);
    float4 v = *(const float4*)(z + (size_t)s * OUT_C + ch);
    float* op = outacc + (size_t)d * OUT_C + ch;
    atomicAdd(op + 0, ex * v.x);
    atomicAdd(op + 1, ex * v.y);
    atomicAdd(op + 2, ex * v.z);
    atomicAdd(op + 3, ex * v.w);
}

// ---------- K7: one wave32 per node: normalize, +b2, log_softmax ----------
__global__ void k_out(const float* __restrict__ outacc, const float* __restrict__ denom,
                      const float* __restrict__ b2, float* __restrict__ out, int n) {
    int node = (blockIdx.x * blockDim.x + threadIdx.x) >> 5;
    int lane = threadIdx.x & 31;
    if (node >= n) return;
    float val = outacc[(size_t)node * OUT_C + lane] / denom[node] + b2[lane];
    float mx = val;
#pragma unroll
    for (int off = 16; off; off >>= 1) mx = fmaxf(mx, __shfl_xor(mx, off, 32));
    float s = __expf(val - mx);
#pragma unroll
    for (int off = 16; off; off >>= 1) s += __shfl_xor(s, off, 32);
    out[(size_t)node * OUT_C + lane] = val - mx - __logf(s);
}

extern "C" void kernel_launch(void* const* d_in, const int* in_sizes, int n_in,
                              void* d_out, int out_size, void* d_ws, size_t ws_size,
                              hipStream_t stream) {
    const float* x     = (const float*)d_in[0];
    const int*   ei    = (const int*)d_in[1];
    const float* Wl    = (const float*)d_in[2];
    const float* Wr    = (const float*)d_in[3];
    const float* b1    = (const float*)d_in[4];
    const float* Wg    = (const float*)d_in[5];
    const float* att_s = (const float*)d_in[6];
    const float* att_d = (const float*)d_in[7];
    const float* b2    = (const float*)d_in[8];
    float* out = (float*)d_out;

    const int n  = in_sizes[0] / IN_C;  // 100000
    const int nE = in_sizes[1] / 2;     // 1600000

    // workspace carve-up (256B aligned)
    char* p = (char*)d_ws;
    auto carve = [&](size_t bytes) {
        char* r = p;
        p += (bytes + 255) & ~(size_t)255;
        return r;
    };
    float*    agg    = (float*)carve((size_t)n * IN_C * 4);    // 25.6 MB
    float*    deg    = (float*)carve((size_t)n * 4);
    float*    z      = (float*)carve((size_t)n * OUT_C * 4);   // 12.8 MB
    float*    a_s    = (float*)carve((size_t)n * 4);
    float*    a_d    = (float*)carve((size_t)n * 4);
    unsigned* m_u    = (unsigned*)carve((size_t)n * 4);
    float*    mbuf   = (float*)carve((size_t)n * 4);
    float*    denom  = (float*)carve((size_t)n * 4);
    float*    outacc = (float*)carve((size_t)n * OUT_C * 4);   // 12.8 MB

    hipMemsetAsync(agg, 0, (size_t)n * IN_C * 4, stream);
    hipMemsetAsync(deg, 0, (size_t)n * 4, stream);

    {   // SAGE scatter
        long long total = (long long)nE * 16;
        int blocks = (int)((total + 255) / 256);
        k_agg<<<blocks, 256, 0, stream>>>(x, ei, agg, deg, nE);
    }
    {   // WMMA transforms
        int blocks = (n + 127) / 128;
        k_gemm<<<blocks, 256, 0, stream>>>(x, agg, deg, Wl, Wr, b1, Wg, z, n);
    }
    k_att<<<(n + 255) / 256, 256, 0, stream>>>(z, att_s, att_d, a_s, a_d, m_u, n);
    k_emax<<<(nE + 255) / 256, 256, 0, stream>>>(ei, a_s, a_d, m_u, nE);
    k_prep<<<(n + 255) / 256, 256, 0, stream>>>(m_u, a_s, a_d, z, mbuf, denom, outacc, n);
    {
        long long total = (long long)nE * 8;
        int blocks = (int)((total + 255) / 256);
        k_eagg<<<blocks, 256, 0, stream>>>(ei, a_s, a_d, mbuf, z, denom, outacc, nE);
    }
    {
        long long total = (long long)n * 32;
        int blocks = (int)((total + 255) / 256);
        k_out<<<blocks, 256, 0, stream>>>(outacc, denom, b2, out, n);
    }
    (void)out_size; (void)ws_size; (void)n_in;
}